// GCBlock_21483426414653
// MI455X (gfx1250) — compile-verified
//
#include <hip/hip_runtime.h>
#include <math.h>

// ---------------------------------------------------------------------------
// CDNA5 WMMA helpers (f32 16x16x4, full-precision path)
// ---------------------------------------------------------------------------
typedef __attribute__((ext_vector_type(2))) float v2f;
typedef __attribute__((ext_vector_type(8))) float v8f;

__device__ __forceinline__ v8f wmma_f32(v2f a, v2f b, v8f c) {
    // D = A(16x4) * B(4x16) + C(16x16), f32 everywhere.
    return __builtin_amdgcn_wmma_f32_16x16x4_f32(
        /*neg_a=*/false, a, /*neg_b=*/false, b,
        /*c_mod=*/(short)0, c, /*reuse_a=*/false, /*reuse_b=*/false);
}

// ---------------------------------------------------------------------------
// Generic WMMA GEMM:  Out[M,NC] = (A[M,K] + abias[K]) * W[NC,K]^T + obias[NC]
// 8 waves/block, each wave computes 16 rows x NC cols. W is staged in LDS in
// 32-wide K slices (feeds ds_load_b64 B-fragments).
// ---------------------------------------------------------------------------
template<int K, int NC>
__global__ __launch_bounds__(256) void gemm_wmma(
    const float* __restrict__ A, const float* __restrict__ W,
    const float* __restrict__ abias, const float* __restrict__ obias,
    float* __restrict__ Out, int M)
{
    constexpr int KT = 32;
    __shared__ float sW[NC * KT];

    const int tid  = threadIdx.x;
    const int wave = tid >> 5;
    const int lane = tid & 31;
    const int half = lane >> 4;     // 0: lanes 0-15, 1: lanes 16-31
    const int lr   = lane & 15;

    const long row0 = (long)blockIdx.x * 128 + (long)wave * 16;

    v8f zero = {};
    v8f acc[NC / 16];
#pragma unroll
    for (int t = 0; t < NC / 16; ++t) acc[t] = zero;

    long arow = row0 + lr;
    if (arow > (long)M - 1) arow = (long)M - 1;   // clamp (EXEC must stay full for WMMA)
    const float* aRow = A + arow * (long)K;

    for (int k0 = 0; k0 < K; k0 += KT) {
        __syncthreads();
        // cooperative stage of W[:, k0:k0+KT) -> sW[n*KT + kk]
        for (int i = tid; i < NC * (KT / 2); i += 256) {
            int n  = i / (KT / 2);
            int kk = (i % (KT / 2)) * 2;
            *reinterpret_cast<float2*>(&sW[n * KT + kk]) =
                *reinterpret_cast<const float2*>(&W[(size_t)n * K + k0 + kk]);
        }
        __syncthreads();
#pragma unroll
        for (int kk = 0; kk < KT; kk += 4) {
            const int kc = kk + 2 * half;
            float2 af = *reinterpret_cast<const float2*>(aRow + k0 + kc);
            if (abias) { af.x += abias[k0 + kc]; af.y += abias[k0 + kc + 1]; }
            v2f a; a[0] = af.x; a[1] = af.y;
#pragma unroll
            for (int nt = 0; nt < NC / 16; ++nt) {
                const float* wp = &sW[(nt * 16 + lr) * KT + kc];
                v2f b; b[0] = wp[0]; b[1] = wp[1];
                acc[nt] = wmma_f32(a, b, acc[nt]);
            }
        }
    }

#pragma unroll
    for (int nt = 0; nt < NC / 16; ++nt) {
        const int col = nt * 16 + lr;
        const float ob = obias ? obias[col] : 0.0f;
#pragma unroll
        for (int r = 0; r < 8; ++r) {
            long row = row0 + r + 8 * half;
            if (row < (long)M)
                Out[(size_t)row * NC + col] = acc[nt][r] + ob;
        }
    }
}

// ---------------------------------------------------------------------------
// Fused attention-logit kernel.
//   e_tile[16,128] = feat[16,32] @ Wc[128,32]^T    (WMMA)
//   u = leaky_relu(xl[src] + xr[dst] + e, 0.2)
//   logits[row,h] = sum_d u[row,h,d] * att[h,d]    (head == WMMA n-tile!)
// srcIdx/dstIdx == nullptr -> identity mapping (self-loop rows).
// ---------------------------------------------------------------------------
__global__ __launch_bounds__(256) void attn_logits(
    const float* __restrict__ feat,      // [rows, 32]
    const float* __restrict__ Wc,        // [128, 32]
    const float* __restrict__ xl,        // [N, 128]
    const float* __restrict__ xr,        // [N, 128]
    const float* __restrict__ att,       // [128] = [8 heads][16]
    const int* __restrict__ srcIdx, const int* __restrict__ dstIdx,
    float* __restrict__ logits,          // write at (rowBase+row)*8 + h
    int rows, long rowBase)
{
    __shared__ float sWc[128 * 32];
    __shared__ float sAtt[128];
    __shared__ int   sSrc[128];
    __shared__ int   sDst[128];

    const int tid  = threadIdx.x;
    const int wave = tid >> 5;
    const int lane = tid & 31;
    const int half = lane >> 4;
    const int lr   = lane & 15;
    const long row0 = (long)blockIdx.x * 128 + (long)wave * 16;

    for (int i = tid; i < 128 * 16; i += 256) {
        int n = i >> 4, kk = (i & 15) * 2;
        *reinterpret_cast<float2*>(&sWc[n * 32 + kk]) =
            *reinterpret_cast<const float2*>(&Wc[n * 32 + kk]);
    }
    if (tid < 128) sAtt[tid] = att[tid];
    if (lane < 16) {
        long rr = row0 + lane;
        if (rr > (long)rows - 1) rr = (long)rows - 1;
        int s, d;
        if (srcIdx) { s = srcIdx[rr]; d = dstIdx[rr]; }
        else        { s = (int)rr;    d = (int)rr; }
        sSrc[wave * 16 + lane] = s;
        sDst[wave * 16 + lane] = d;
    }
    __syncthreads();

    long ar = row0 + lr;
    if (ar > (long)rows - 1) ar = (long)rows - 1;
    const float* fRow = feat + ar * 32;

    v8f zero = {};
    v8f acc[8];
#pragma unroll
    for (int h = 0; h < 8; ++h) acc[h] = zero;

#pragma unroll
    for (int kk = 0; kk < 32; kk += 4) {
        const int kc = kk + 2 * half;
        float2 af = *reinterpret_cast<const float2*>(fRow + kc);
        v2f a; a[0] = af.x; a[1] = af.y;
#pragma unroll
        for (int h = 0; h < 8; ++h) {
            const float* wp = &sWc[(h * 16 + lr) * 32 + kc];
            v2f b; b[0] = wp[0]; b[1] = wp[1];
            acc[h] = wmma_f32(a, b, acc[h]);
        }
    }

#pragma unroll
    for (int h = 0; h < 8; ++h) {
        const float av = sAtt[h * 16 + lr];
#pragma unroll
        for (int r = 0; r < 8; ++r) {
            const int li = wave * 16 + r + 8 * half;
            const long row = row0 + r + 8 * half;
            const int s = sSrc[li], d = sDst[li];
            const int c = h * 16 + lr;
            float v = acc[h][r] + xl[(size_t)s * 128 + c] + xr[(size_t)d * 128 + c];
            v = v > 0.0f ? v : 0.2f * v;
            float t = v * av;
            t += __shfl_xor(t, 1, 32);
            t += __shfl_xor(t, 2, 32);
            t += __shfl_xor(t, 4, 32);
            t += __shfl_xor(t, 8, 32);   // sum over the head's 16 columns
            if (lr == 0 && row < (long)rows)
                logits[(size_t)(rowBase + row) * 8 + h] = t;
        }
    }
}

// ---------------------------------------------------------------------------
// Small helper kernels
// ---------------------------------------------------------------------------
__global__ void fillf(float* __restrict__ p, float v, long n) {
    long i = (long)blockIdx.x * blockDim.x + threadIdx.x;
    if (i < n) p[i] = v;
}

// Wc[c][k] = sum_j We[c][j] * W_r2m[j][k]   (128x32, K=128)
__global__ void compute_wc(const float* __restrict__ We,
                           const float* __restrict__ Wr2m,
                           float* __restrict__ Wc) {
    int idx = blockIdx.x * 256 + threadIdx.x;
    if (idx >= 128 * 32) return;
    int c = idx >> 5, k = idx & 31;
    float s = 0.0f;
    for (int j = 0; j < 128; ++j) s = fmaf(We[c * 128 + j], Wr2m[j * 32 + k], s);
    Wc[idx] = s;
}

// per-edge: rbfsum[dst] += rbf[e];  deg[dst] += 1
__global__ void rbf_scatter(const float* __restrict__ rbf, const int* __restrict__ dst,
                            float* __restrict__ rbfsum, float* __restrict__ deg, long E) {
    long i = (long)blockIdx.x * blockDim.x + threadIdx.x;
    long e = i >> 5; int k = (int)(i & 31);
    if (e >= E) return;
    int d = dst[e];
    atomicAdd(&rbfsum[(size_t)d * 32 + k], rbf[(size_t)e * 32 + k]);
    if (k == 0) atomicAdd(&deg[d], 1.0f);
}

__global__ void rbf_mean(float* __restrict__ rbfsum, const float* __restrict__ deg, long n32) {
    long i = (long)blockIdx.x * blockDim.x + threadIdx.x;
    if (i >= n32) return;
    long n = i >> 5;
    rbfsum[i] /= fmaxf(deg[n], 1.0f);
}

__device__ __forceinline__ void atomicMaxF(float* addr, float v) {
    if (v >= 0.0f) atomicMax((int*)addr, __float_as_int(v));
    else           atomicMin((unsigned int*)addr, __float_as_uint(v));
}

__global__ void seg_max(const float* __restrict__ logits, const int* __restrict__ dstIdx,
                        float* __restrict__ nmax, long E, long total) {
    long i = (long)blockIdx.x * blockDim.x + threadIdx.x;
    if (i >= total * 8) return;
    long row = i >> 3; int h = (int)(i & 7);
    long d = (row < E) ? (long)dstIdx[row] : (row - E);
    atomicMaxF(&nmax[d * 8 + h], logits[i]);
}

__global__ void seg_expsum(float* __restrict__ logits, const int* __restrict__ dstIdx,
                           const float* __restrict__ nmax, float* __restrict__ denom,
                           long E, long total) {
    long i = (long)blockIdx.x * blockDim.x + threadIdx.x;
    if (i >= total * 8) return;
    long row = i >> 3; int h = (int)(i & 7);
    long d = (row < E) ? (long)dstIdx[row] : (row - E);
    float p = __expf(logits[i] - nmax[d * 8 + h]);
    logits[i] = p;                       // overwrite logit with exp-numerator
    atomicAdd(&denom[d * 8 + h], p);
}

// one wave per row: agg[dst] += xl[src] * alpha[row,h]
__global__ __launch_bounds__(256) void aggregate(
    const float* __restrict__ xl, const float* __restrict__ p,
    const float* __restrict__ denom,
    const int* __restrict__ srcIdx, const int* __restrict__ dstIdx,
    float* __restrict__ agg, long E, long total)
{
    long wid = ((long)blockIdx.x * blockDim.x + threadIdx.x) >> 5;
    int lane = threadIdx.x & 31;
    if (wid >= total) return;
    long s, d;
    if (wid < E) { s = srcIdx[wid]; d = dstIdx[wid]; }
    else         { s = wid - E;     d = wid - E; }
    const int h = lane >> 2;                       // 4 cols per lane, head-aligned
    const float alpha = p[(size_t)wid * 8 + h] / denom[(size_t)d * 8 + h];
    const float4 v = *reinterpret_cast<const float4*>(&xl[(size_t)s * 128 + lane * 4]);
    float* ap = &agg[(size_t)d * 128 + lane * 4];
    atomicAdd(ap + 0, v.x * alpha);
    atomicAdd(ap + 1, v.y * alpha);
    atomicAdd(ap + 2, v.z * alpha);
    atomicAdd(ap + 3, v.w * alpha);
}

// ---------------------------------------------------------------------------
// Host-side orchestration
// ---------------------------------------------------------------------------
extern "C" void kernel_launch(void* const* d_in, const int* in_sizes, int n_in,
                              void* d_out, int out_size, void* d_ws, size_t ws_size,
                              hipStream_t stream) {
    const float* x     = (const float*)d_in[0];
    const float* rbf   = (const float*)d_in[1];
    const int*   eidx  = (const int*)d_in[2];
    // d_in[3] = batch (unused)
    const float* W_n2m = (const float*)d_in[4];
    const float* W_r2m = (const float*)d_in[5];
    const float* Wl    = (const float*)d_in[6];
    const float* bl    = (const float*)d_in[7];
    const float* Wr    = (const float*)d_in[8];
    const float* br    = (const float*)d_in[9];
    const float* We    = (const float*)d_in[10];
    const float* att   = (const float*)d_in[11];
    const float* bias  = (const float*)d_in[12];
    const float* W_m2n = (const float*)d_in[13];
    float* out = (float*)d_out;
    (void)n_in; (void)out_size; (void)ws_size;

    const long N = (long)in_sizes[0] / 256;
    const long E = (long)in_sizes[1] / 32;
    const int* srcIdx = eidx;
    const int* dstIdx = eidx + E;

    // workspace layout (floats)
    float* w       = (float*)d_ws;
    float* xm      = w;  w += N * 128;
    float* xl      = w;  w += N * 128;
    float* xr      = w;  w += N * 128;
    float* agg     = w;  w += N * 128;
    float* rbfsum  = w;  w += N * 32;      // becomes rbf_mean in place
    float* deg     = w;  w += N;
    float* nmax    = w;  w += N * 8;
    float* denom   = w;  w += N * 8;
    float* logits  = w;  w += (E + N) * 8; // becomes exp-numerators in place
    float* Wc      = w;  w += 128 * 32;

    auto cdiv = [](long a, long b) { return (int)((a + b - 1) / b); };

    // init accumulators
    fillf<<<cdiv(N, 256), 256, 0, stream>>>(deg, 0.0f, N);
    fillf<<<cdiv(N * 32, 256), 256, 0, stream>>>(rbfsum, 0.0f, N * 32);
    fillf<<<cdiv(N * 8, 256), 256, 0, stream>>>(nmax, -INFINITY, N * 8);
    fillf<<<cdiv(N * 8, 256), 256, 0, stream>>>(denom, 0.0f, N * 8);
    fillf<<<cdiv(N * 128, 256), 256, 0, stream>>>(agg, 0.0f, N * 128);

    // fold the two edge linear maps: Wc = We @ W_r2m
    compute_wc<<<16, 256, 0, stream>>>(We, W_r2m, Wc);

    // node transforms (WMMA GEMMs)
    const int gb = cdiv(N, 128);
    gemm_wmma<256, 128><<<gb, 256, 0, stream>>>(x,  W_n2m, nullptr, nullptr, xm, (int)N);
    gemm_wmma<128, 128><<<gb, 256, 0, stream>>>(xm, Wl,    nullptr, bl,      xl, (int)N);
    gemm_wmma<128, 128><<<gb, 256, 0, stream>>>(xm, Wr,    nullptr, br,      xr, (int)N);

    // self-loop attrs via rbf sums (32-dim atomics instead of 128-dim)
    rbf_scatter<<<cdiv(E * 32, 256), 256, 0, stream>>>(rbf, dstIdx, rbfsum, deg, E);
    rbf_mean<<<cdiv(N * 32, 256), 256, 0, stream>>>(rbfsum, deg, N * 32);

    // fused edge-transform + attention logits (WMMA)
    attn_logits<<<cdiv(E, 128), 256, 0, stream>>>(rbf,    Wc, xl, xr, att, srcIdx, dstIdx,
                                                  logits, (int)E, 0L);
    attn_logits<<<cdiv(N, 128), 256, 0, stream>>>(rbfsum, Wc, xl, xr, att, nullptr, nullptr,
                                                  logits, (int)N, E);

    // segment softmax (max, exp, sum)
    const long total = E + N;
    seg_max<<<cdiv(total * 8, 256), 256, 0, stream>>>(logits, dstIdx, nmax, E, total);
    seg_expsum<<<cdiv(total * 8, 256), 256, 0, stream>>>(logits, dstIdx, nmax, denom, E, total);

    // weighted scatter of xl[src] into agg[dst]
    aggregate<<<cdiv(total * 32, 256), 256, 0, stream>>>(xl, logits, denom, srcIdx, dstIdx,
                                                         agg, E, total);

    // final projection: out = (agg + bias) @ W_m2n^T  (bias folded into A-fragment)
    gemm_wmma<128, 256><<<gb, 256, 0, stream>>>(agg, W_m2n, bias, nullptr, out, (int)N);
}